// MLP_mia_white2_m_65300682768665
// MI455X (gfx1250) — compile-verified
//
#include <hip/hip_runtime.h>
#include <hip/hip_bf16.h>

// Problem constants (match reference)
#define T_DIM   8
#define N_NODES 100000
#define D_DIM   128
#define E_EDGES 50000
#define MS_DIM  4
#define H0_DIM  64
#define H1_DIM  32
#define EPS_F   1e-8f

typedef __attribute__((ext_vector_type(2))) float v2f;
typedef __attribute__((ext_vector_type(8))) float v8f;

// ---------------------------------------------------------------------------
// Pass 1: streaming norms.  norms[(t*MS+m)*N + n] = sqrt(sum_d z1[t,n,d]^2 * w1[m,d]^2)
// One wave per (t,n) row; fully coalesced read of the 410MB tensor exactly once.
// ---------------------------------------------------------------------------
__global__ void norms_kernel(const float* __restrict__ z1,
                             const float* __restrict__ w1,
                             float* __restrict__ norms) {
    __shared__ float wsq[MS_DIM * D_DIM];
    for (int i = threadIdx.x; i < MS_DIM * D_DIM; i += blockDim.x) {
        float w = w1[i];
        wsq[i] = w * w;
    }
    __syncthreads();

    const int lane = threadIdx.x & 31;
    const int wid  = threadIdx.x >> 5;
    const int wpb  = blockDim.x >> 5;
    const int numRows = T_DIM * N_NODES;

    for (int row = blockIdx.x * wpb + wid; row < numRows; row += gridDim.x * wpb) {
        const float4 v = *(const float4*)(z1 + (size_t)row * D_DIM + lane * 4);
        const int k = lane * 4;
        float vv[4] = { v.x * v.x, v.y * v.y, v.z * v.z, v.w * v.w };
        float s0 = 0.f, s1 = 0.f, s2 = 0.f, s3 = 0.f;
#pragma unroll
        for (int i = 0; i < 4; ++i) {
            s0 += vv[i] * wsq[0 * D_DIM + k + i];
            s1 += vv[i] * wsq[1 * D_DIM + k + i];
            s2 += vv[i] * wsq[2 * D_DIM + k + i];
            s3 += vv[i] * wsq[3 * D_DIM + k + i];
        }
#pragma unroll
        for (int off = 16; off >= 1; off >>= 1) {
            s0 += __shfl_xor(s0, off, 32);
            s1 += __shfl_xor(s1, off, 32);
            s2 += __shfl_xor(s2, off, 32);
            s3 += __shfl_xor(s3, off, 32);
        }
        if (lane == 0) {
            const int t = row / N_NODES;
            const int n = row - t * N_NODES;
            norms[(size_t)(t * MS_DIM + 0) * N_NODES + n] = sqrtf(s0);
            norms[(size_t)(t * MS_DIM + 1) * N_NODES + n] = sqrtf(s1);
            norms[(size_t)(t * MS_DIM + 2) * N_NODES + n] = sqrtf(s2);
            norms[(size_t)(t * MS_DIM + 3) * N_NODES + n] = sqrtf(s3);
        }
    }
}

// ---------------------------------------------------------------------------
// Pass 2: per-edge weighted dot via V_WMMA_F32_16X16X4_F32.
// One wave handles a tile of 16 edges (E = 3125 * 16 exactly, no partial tiles).
// A (16x4 slice) = elementwise a*b for 16 edges; B (4x16) = squared weight rows
// in columns 0..3.  Accumulate over K = 128 in chunks of 4.  C[edge][m] = dot.
// Fragment placement per ISA 7.12.2:
//   A: lanes 0-15 rows M, {vgpr0,vgpr1} = K{0,1}; lanes 16-31 = K{2,3}
//   B: mirrored (lanes = columns, K split across lane halves)
//   C: lane = column (m), vgpr j = row j (lanes 0-15) / row j+8 (lanes 16-31)
// ---------------------------------------------------------------------------
__global__ void edge_sim_kernel(const int* __restrict__ edge_list,
                                const float* __restrict__ z1,
                                const float* __restrict__ w1,
                                const float* __restrict__ norms,
                                float* __restrict__ sim) {
    __shared__ float wsq[MS_DIM * D_DIM];
    for (int i = threadIdx.x; i < MS_DIM * D_DIM; i += blockDim.x) {
        float w = w1[i];
        wsq[i] = w * w;
    }
    __syncthreads();   // before any wave-uniform early exit (EXEC stays full)

    const int lane  = threadIdx.x & 31;
    const int wid   = threadIdx.x >> 5;
    const int tile  = blockIdx.x * (blockDim.x >> 5) + wid;
    const int numTiles = E_EDGES / 16;            // 3125, exact
    if (tile >= numTiles) return;                  // wave-uniform

    const int base  = tile * 16;
    const int row   = lane & 15;                   // A-matrix row (edge in tile)
    const int khalf = lane >> 4;                   // 0 -> K{0,1}, 1 -> K{2,3}

    const int srcR = edge_list[2 * (base + row) + 0];
    const int dstR = edge_list[2 * (base + row) + 1];

    const int m = row;                             // C column; weight head if < 4
    const float mMask = (m < MS_DIM) ? 1.0f : 0.0f;
    const int mC = (m < MS_DIM) ? m : 0;           // clamped LDS index

    for (int t = 0; t < T_DIM; ++t) {
        const float* aRow = z1 + ((size_t)t * N_NODES + srcR) * D_DIM;
        const float* bRow = z1 + ((size_t)t * N_NODES + dstR) * D_DIM;

        v8f acc = { 0.f, 0.f, 0.f, 0.f, 0.f, 0.f, 0.f, 0.f };
        for (int c = 0; c < D_DIM / 4; ++c) {
            const int kb = c * 4 + khalf * 2;
            const v2f av = *(const v2f*)(aRow + kb);
            const v2f bv = *(const v2f*)(bRow + kb);
            v2f afrag;
            afrag.x = av.x * bv.x;
            afrag.y = av.y * bv.y;
            v2f bfrag;
            bfrag.x = wsq[mC * D_DIM + kb]     * mMask;
            bfrag.y = wsq[mC * D_DIM + kb + 1] * mMask;
            acc = __builtin_amdgcn_wmma_f32_16x16x4_f32(
                      false, afrag, false, bfrag, (short)0, acc, false, false);
        }

        // Finish cosines and reduce the 4 head-lanes per edge.
#pragma unroll
        for (int j = 0; j < 8; ++j) {
            const int e = base + khalf * 8 + j;    // edge held in acc[j] on this lane
            float cosv = 0.f;
            if (m < MS_DIM) {
                const int s = edge_list[2 * e + 0];
                const int d = edge_list[2 * e + 1];
                const float na = norms[(size_t)(t * MS_DIM + m) * N_NODES + s];
                const float nb = norms[(size_t)(t * MS_DIM + m) * N_NODES + d];
                cosv = acc[j] / (fmaxf(na, EPS_F) * fmaxf(nb, EPS_F));
            }
            // butterfly over lanes {0..3} / {16..19} (and harmlessly over the rest)
            cosv += __shfl_xor(cosv, 1, 32);
            cosv += __shfl_xor(cosv, 2, 32);
            if ((lane & 15) == 0)
                sim[(size_t)e * T_DIM + t] = cosv * (1.0f / MS_DIM);
        }
    }
}

// ---------------------------------------------------------------------------
// Pass 3: MLP 8 -> 64 -> 32 -> 1 over 50k rows.  Weights cached in LDS; h0 is
// streamed one neuron at a time into 32 live h1 accumulators (no spills).
// ---------------------------------------------------------------------------
__global__ void mlp_kernel(const float* __restrict__ sim,
                           const float* __restrict__ W0, const float* __restrict__ b0,
                           const float* __restrict__ W1, const float* __restrict__ b1,
                           const float* __restrict__ Wp, const float* __restrict__ bp,
                           float* __restrict__ out) {
    __shared__ float sW0[H0_DIM * T_DIM];
    __shared__ float sB0[H0_DIM];
    __shared__ float sW1[H1_DIM * H0_DIM];
    __shared__ float sB1[H1_DIM];
    __shared__ float sWp[H1_DIM];
    __shared__ float sBp;
    for (int i = threadIdx.x; i < H0_DIM * T_DIM; i += blockDim.x) sW0[i] = W0[i];
    for (int i = threadIdx.x; i < H0_DIM;         i += blockDim.x) sB0[i] = b0[i];
    for (int i = threadIdx.x; i < H1_DIM * H0_DIM; i += blockDim.x) sW1[i] = W1[i];
    for (int i = threadIdx.x; i < H1_DIM;         i += blockDim.x) sB1[i] = b1[i];
    for (int i = threadIdx.x; i < H1_DIM;         i += blockDim.x) sWp[i] = Wp[i];
    if (threadIdx.x == 0) sBp = bp[0];
    __syncthreads();

    const int e = blockIdx.x * blockDim.x + threadIdx.x;
    if (e >= E_EDGES) return;

    float x[T_DIM];
    {
        const float4 x0 = *(const float4*)(sim + (size_t)e * T_DIM);
        const float4 x1 = *(const float4*)(sim + (size_t)e * T_DIM + 4);
        x[0] = x0.x; x[1] = x0.y; x[2] = x0.z; x[3] = x0.w;
        x[4] = x1.x; x[5] = x1.y; x[6] = x1.z; x[7] = x1.w;
    }

    float h1[H1_DIM];
#pragma unroll
    for (int i = 0; i < H1_DIM; ++i) h1[i] = sB1[i];

    for (int j = 0; j < H0_DIM; ++j) {
        float s = sB0[j];
#pragma unroll
        for (int k = 0; k < T_DIM; ++k) s += x[k] * sW0[j * T_DIM + k];
        s = fmaxf(s, 0.f);
#pragma unroll
        for (int i = 0; i < H1_DIM; ++i) h1[i] += s * sW1[i * H0_DIM + j];
    }

    float p = sBp;
#pragma unroll
    for (int i = 0; i < H1_DIM; ++i) p += fmaxf(h1[i], 0.f) * sWp[i];

    out[e] = p;
}

// ---------------------------------------------------------------------------
// kernel_launch: inputs per setup_inputs() order.
//   0 edge_list (E,2) int   1 z1 (T,N,D) f32   2 z2 (unused)
//   3 w1 (MS,D) f32         4 w2 (unused)
//   5 W0 (64,8)  6 b0 (64)  7 W1 (32,64)  8 b1 (32)  9 Wp (1,32)  10 bp (1)
// Workspace: norms (T*MS*N f32 = 12.8 MB) then sim (E*T f32 = 1.6 MB).
// ---------------------------------------------------------------------------
extern "C" void kernel_launch(void* const* d_in, const int* in_sizes, int n_in,
                              void* d_out, int out_size, void* d_ws, size_t ws_size,
                              hipStream_t stream) {
    const int*   edge_list = (const int*)d_in[0];
    const float* z1 = (const float*)d_in[1];
    const float* w1 = (const float*)d_in[3];
    const float* W0 = (const float*)d_in[5];
    const float* b0 = (const float*)d_in[6];
    const float* W1 = (const float*)d_in[7];
    const float* b1 = (const float*)d_in[8];
    const float* Wp = (const float*)d_in[9];
    const float* bp = (const float*)d_in[10];

    float* norms = (float*)d_ws;                                   // T*MS*N floats
    float* sim   = norms + (size_t)T_DIM * MS_DIM * N_NODES;        // E*T floats
    float* out   = (float*)d_out;

    // Pass 1: streaming norms (grid-stride, 8 waves/block).
    norms_kernel<<<4096, 256, 0, stream>>>(z1, w1, norms);

    // Pass 2: 3125 edge tiles, 8 waves/block -> 391 blocks.
    const int numTiles = E_EDGES / 16;
    const int blocks2  = (numTiles + 7) / 8;
    edge_sim_kernel<<<blocks2, 256, 0, stream>>>(edge_list, z1, w1, norms, sim);

    // Pass 3: MLP, one thread per edge.
    const int blocks3 = (E_EDGES + 255) / 256;
    mlp_kernel<<<blocks3, 256, 0, stream>>>(sim, W0, b0, W1, b1, Wp, bp, out);
}